// MixConv_90872918049350
// MI455X (gfx1250) — compile-verified
//
#include <hip/hip_runtime.h>
#include <hip/hip_bf16.h>

typedef __attribute__((ext_vector_type(2))) float v2f;
typedef __attribute__((ext_vector_type(8))) float v8f;

#define BATCH 16
#define FIN   16
#define FOUT  16
#define HW    (384 * 384)          // 147456 pixels per (b, channel) plane
#define LAT   512
#define NMIX  8

#define TILES_PER_WAVE   8
#define WAVES_PER_BLOCK  8
#define TILES_PER_BLOCK  (TILES_PER_WAVE * WAVES_PER_BLOCK)   // 64
#define TILES_PER_BATCH  (HW / 16)                            // 9216
#define BLOCKS_PER_BATCH (TILES_PER_BATCH / TILES_PER_BLOCK)  // 144

// ---------------------------------------------------------------------------
// Stage 1: tiny precompute of per-sample dynamic kernel + bias into workspace.
//   mix[b,m]   = lat[b,:] . w_dyn[m,:] + b_dyn[m]          (16x8)
//   kern[b,o,i]= sum_m mix[b,m] * kernel_mix[m,o,i]        (16x16x16)
//   bias[b,o]  = sum_m mix[b,m] * bias_mix[m,o]            (16x16)
// Single block, 256 threads. Output lands in L2 and is reused by stage 2.
// ---------------------------------------------------------------------------
__global__ void mixconv_precompute(const float* __restrict__ lat,
                                   const float* __restrict__ kernel_mix,
                                   const float* __restrict__ bias_mix,
                                   const float* __restrict__ w_dyn,
                                   const float* __restrict__ b_dyn,
                                   float* __restrict__ kern,
                                   float* __restrict__ bias) {
    __shared__ float mix[BATCH * NMIX];
    const int t = threadIdx.x;

    if (t < BATCH * NMIX) {
        const int b = t / NMIX;
        const int m = t % NMIX;
        float acc = b_dyn[m];
        const float* lb = lat + b * LAT;
        const float* wm = w_dyn + m * LAT;
        #pragma unroll 4
        for (int l = 0; l < LAT; ++l) acc += lb[l] * wm[l];
        mix[t] = acc;
    }
    __syncthreads();

    // dynamic 1x1 kernel: 4096 entries
    for (int idx = t; idx < BATCH * FOUT * FIN; idx += 256) {
        const int b  = idx >> 8;        // / (FOUT*FIN)
        const int oi = idx & 255;
        float acc = 0.0f;
        #pragma unroll
        for (int m = 0; m < NMIX; ++m)
            acc += mix[b * NMIX + m] * kernel_mix[m * (FOUT * FIN) + oi];
        kern[idx] = acc;
    }

    // dynamic bias: 256 entries, one per thread
    {
        const int b = t >> 4;
        const int o = t & 15;
        float acc = 0.0f;
        #pragma unroll
        for (int m = 0; m < NMIX; ++m)
            acc += mix[b * NMIX + m] * bias_mix[m * FOUT + o];
        bias[b * FOUT + o] = acc;
    }
}

// ---------------------------------------------------------------------------
// Stage 2: streaming per-sample 16x16 matmul over pixels via WMMA f32 16x16x4.
//   D(16 fout x 16 pix) = sum over 4 K-steps A(16x4 of kern) * B(4x16 of x) + C(bias)
// A-frag  (ISA 16x4 f32 layout): lane L, vgpr j -> A(m = L%16, k = 2*(L/16)+j)
// B-frag  (mirror):              lane L, vgpr j -> B(k = 2*(L/16)+j, n = L%16)
// C/D     (ISA 16x16 f32):       lane L, vgpr v -> (m = v + 8*(L/16), n = L%16)
// ---------------------------------------------------------------------------
__global__ void __launch_bounds__(256)
mixconv_main(const float* __restrict__ x,
             const float* __restrict__ kern,
             const float* __restrict__ bias,
             float* __restrict__ out) {
    const int lane = threadIdx.x & 31;
    const int wave = threadIdx.x >> 5;
    const int half = lane >> 4;     // 0: lanes 0-15, 1: lanes 16-31
    const int lm   = lane & 15;

    const int b     = blockIdx.x / BLOCKS_PER_BATCH;
    const int bib   = blockIdx.x % BLOCKS_PER_BATCH;
    const int tile0 = bib * TILES_PER_BLOCK + wave * TILES_PER_WAVE;

    // Preload the four A fragments (K-steps 0..3) for this sample: kern[b, m, k]
    const float* kb = kern + (b * FOUT + lm) * FIN;   // row m = lm
    v2f a[4];
    #pragma unroll
    for (int ks = 0; ks < 4; ++ks) {
        const int k0 = ks * 4 + 2 * half;
        a[ks].x = kb[k0 + 0];
        a[ks].y = kb[k0 + 1];
    }

    // Bias-seeded accumulator template: c[v] holds row m = v + 8*half
    v8f cinit;
    #pragma unroll
    for (int v = 0; v < 8; ++v) cinit[v] = bias[b * FOUT + v + 8 * half];

    const float* xb = x   + (size_t)b * FIN  * HW;
    float*       ob = out + (size_t)b * FOUT * HW;

    #pragma unroll
    for (int t = 0; t < TILES_PER_WAVE; ++t) {
        const int p = (tile0 + t) * 16 + lm;   // this lane's pixel column

        v8f c = cinit;
        #pragma unroll
        for (int ks = 0; ks < 4; ++ks) {
            const int k0 = ks * 4 + 2 * half;
            v2f bf;
            bf.x = xb[(size_t)(k0 + 0) * HW + p];
            bf.y = xb[(size_t)(k0 + 1) * HW + p];
            // 8 args: (neg_a, A, neg_b, B, c_mod, C, reuse_a, reuse_b)
            c = __builtin_amdgcn_wmma_f32_16x16x4_f32(
                    false, a[ks], false, bf, (short)0, c, false, false);
        }

        #pragma unroll
        for (int v = 0; v < 8; ++v)
            ob[(size_t)(v + 8 * half) * HW + p] = c[v];
    }
}

extern "C" void kernel_launch(void* const* d_in, const int* in_sizes, int n_in,
                              void* d_out, int out_size, void* d_ws, size_t ws_size,
                              hipStream_t stream) {
    const float* x          = (const float*)d_in[0];
    const float* lat        = (const float*)d_in[1];
    const float* kernel_mix = (const float*)d_in[2];
    const float* bias_mix   = (const float*)d_in[3];
    const float* w_dyn      = (const float*)d_in[4];
    const float* b_dyn      = (const float*)d_in[5];
    float*       out        = (float*)d_out;

    float* kern_ws = (float*)d_ws;                       // 16*16*16 floats
    float* bias_ws = kern_ws + BATCH * FOUT * FIN;       // 16*16 floats

    mixconv_precompute<<<1, 256, 0, stream>>>(lat, kernel_mix, bias_mix,
                                              w_dyn, b_dyn, kern_ws, bias_ws);

    const int grid = BATCH * BLOCKS_PER_BATCH;           // 2304 blocks
    mixconv_main<<<grid, 256, 0, stream>>>(x, kern_ws, bias_ws, out);
}